// Hamiltonian_14972255994466
// MI455X (gfx1250) — compile-verified
//
#include <hip/hip_runtime.h>
#include <hip/hip_bf16.h>

#define NMOLC   256
#define MS      48
#define NATOMS  (NMOLC*MS)              // 12288
#define NORB    (4*MS)                  // 192
#define NPAIRS  (NMOLC*MS*(MS-1)/2)     // 288768
#define MATELEMS ((size_t)NMOLC*NORB*NORB)  // 9437184
#define EVC     27.21f
#define SCF_ITERS 8
#define JAC_SWEEPS 8
// TDM-padded LDS layout: pad of 1 dword after every 64 dwords streamed.
// Element (r,c) of a 192-wide row lives at dword offset r*195 + c + (c>>6).
#define AROW    195
#define AELEMS  37504                   // >= 192*195+3, rounded up

typedef float v2f __attribute__((ext_vector_type(2)));
typedef float v8f __attribute__((ext_vector_type(8)));
typedef unsigned int u32x4 __attribute__((ext_vector_type(4)));
typedef int i32x8 __attribute__((ext_vector_type(8)));
typedef int i32x4 __attribute__((ext_vector_type(4)));

__device__ __forceinline__ int aoff(int r, int c) { return r * AROW + c + (c >> 6); }

// Issue one Tensor Data Mover DMA: 2D f32 tile (rows x cols) global -> LDS,
// with optional 1-dword-per-64-dwords LDS padding. D# per CDNA5 ISA §8.3/8.4.
// clang-23 toolchain: 6-arg builtin (g0, g1, g2, g3, g4, cpol).
__device__ __forceinline__ void tdm_load_f32_2d(const float* gsrc, unsigned lds_off,
                                                int rows, int cols, int row_stride,
                                                int pad_on)
{
    unsigned long long ga = (unsigned long long)(uintptr_t)gsrc;
    u32x4 g0;
    g0[0] = 1u;                                            // count=1, user mode
    g0[1] = lds_off;                                       // lds_addr (bytes)
    g0[2] = (unsigned)ga;                                  // global_addr[31:0]
    g0[3] = ((unsigned)(ga >> 32) & 0x1FFFFFFu) | (2u << 30); // ga[56:32] | type=2
    unsigned dw0 = (2u << 16);                             // data_size = 4 bytes
    if (pad_on) dw0 |= (1u << 20) | (5u << 22);            // pad_enable, interval=64dw, amount=1dw
    unsigned t1s = (unsigned)(rows * row_stride);          // dim1 stride (unused for 2D tile)
    i32x8 g1;
    g1[0] = (int)dw0;                                      // wg_mask=0, flags
    g1[1] = (int)((unsigned)cols << 16);                   // tensor_dim0[15:0]
    g1[2] = (int)(((unsigned)cols >> 16) | ((unsigned)rows << 16)); // dim0 hi | tensor_dim1 lo
    g1[3] = (int)(((unsigned)rows >> 16) | ((unsigned)cols << 16)); // dim1 hi | tile_dim0
    g1[4] = (int)((unsigned)rows & 0xFFFFu);               // tile_dim1 | tile_dim2=0
    g1[5] = (int)(unsigned)row_stride;                     // tensor_dim0_stride[31:0]
    g1[6] = (int)((t1s & 0xFFFFu) << 16);                  // d0s hi=0 | d1s[15:0]
    g1[7] = (int)(t1s >> 16);                              // d1s[47:16]
    i32x4 z4 = { 0, 0, 0, 0 };
    i32x8 z8 = { 0, 0, 0, 0, 0, 0, 0, 0 };
    __builtin_amdgcn_tensor_load_to_lds(g0, g1, z4, z4, z8, 0);
}

// ---------------------------------------------------------------- utilities
__global__ void zero_f32_kernel(float* __restrict__ p, int n) {
    int i = blockIdx.x * blockDim.x + threadIdx.x;
    if (i < n) p[i] = 0.0f;
}

// ---------------------------------------------------------------- Hcore / w
__global__ void pair_hcore_kernel(
    const int* __restrict__ idxi, const int* __restrict__ idxj,
    const int* __restrict__ pair_molid, const int* __restrict__ Z,
    const float* __restrict__ xij, const float* __restrict__ rij,
    const float* __restrict__ zeta_s, const float* __restrict__ zeta_p,
    const float* __restrict__ g_ss,
    const float* __restrict__ beta_s, const float* __restrict__ beta_p,
    float* __restrict__ w_out, float* __restrict__ w_ss,
    float* __restrict__ vatt, float* __restrict__ Hcore)
{
    int p = blockIdx.x * blockDim.x + threadIdx.x;
    if (p >= NPAIRS) return;
    int i = idxi[p], j = idxj[p];
    float r = rij[p];
    float rho = 7.0f / g_ss[i] + 7.0f / g_ss[j];
    float base = 1.0f / sqrtf(r * r + rho * rho);
    float wss = EVC * base;
    w_ss[p] = wss;
    float* wp = w_out + (size_t)p * 100;
#pragma unroll
    for (int a = 0; a < 10; ++a) {
        float sa = 1.0f + 0.05f * (float)a;
#pragma unroll
        for (int b = 0; b < 10; ++b) {
            float sb = 1.0f + 0.05f * (float)b;
            wp[a * 10 + b] = wss * sa * sb;
        }
    }
    float torei = (float)(Z[i] - 2), torej = (float)(Z[j] - 2);
    atomicAdd(&vatt[i], -torej * wss);
    atomicAdd(&vatt[j], -torei * wss);

    float x[3] = { xij[p * 3 + 0], xij[p * 3 + 1], xij[p * 3 + 2] };
    float e_ss = expf(-0.5f * (zeta_s[i] + zeta_s[j]) * r);
    float e_sp = expf(-0.5f * (zeta_s[i] + zeta_p[j]) * r);
    float e_ps = expf(-0.5f * (zeta_p[i] + zeta_s[j]) * r);
    float e_pp = expf(-0.5f * (zeta_p[i] + zeta_p[j]) * r);
    float S[4][4];
    S[0][0] = e_ss;
#pragma unroll
    for (int l = 0; l < 3; ++l) { S[0][1 + l] = x[l] * e_sp; S[1 + l][0] = -x[l] * e_ps; }
#pragma unroll
    for (int k = 0; k < 3; ++k)
#pragma unroll
        for (int l = 0; l < 3; ++l)
            S[1 + k][1 + l] = (((k == l) ? 1.0f : 0.0f) - x[k] * x[l]) * e_pp;
    float boi[4] = { beta_s[i], beta_p[i], beta_p[i], beta_p[i] };
    float boj[4] = { beta_s[j], beta_p[j], beta_p[j], beta_p[j] };
    int mol = pair_molid[p];
    int ai = i % MS, aj = j % MS;
    float* Hm = Hcore + (size_t)mol * NORB * NORB;
#pragma unroll
    for (int u = 0; u < 4; ++u)
#pragma unroll
        for (int v = 0; v < 4; ++v) {
            float ob = 0.5f * (boi[u] + boj[v]) * S[u][v];
            Hm[(4 * ai + u) * NORB + 4 * aj + v] = ob;
            Hm[(4 * aj + v) * NORB + 4 * ai + u] = ob;
        }
}

__global__ void hcore_diag_kernel(
    const int* __restrict__ Z, const int* __restrict__ atom_molid,
    const float* __restrict__ U_ss, const float* __restrict__ U_pp,
    const float* __restrict__ vatt, float* __restrict__ Hcore)
{
    int a = blockIdx.x * blockDim.x + threadIdx.x;
    if (a >= NATOMS) return;
    int mol = atom_molid[a];
    int ai = a % MS;
    float va = vatt[a];
    float ud[4] = { U_ss[a] + va, U_pp[a] + va, U_pp[a] + va, U_pp[a] + va };
    float* Hm = Hcore + (size_t)mol * NORB * NORB;
#pragma unroll
    for (int u = 0; u < 4; ++u)
#pragma unroll
        for (int v = 0; v < 4; ++v)
            Hm[(4 * ai + u) * NORB + 4 * ai + v] = (u == v) ? ud[u] : 0.0f;
}

__global__ void init_P_kernel(const int* __restrict__ Z, float* __restrict__ P)
{
    size_t idx = (size_t)blockIdx.x * blockDim.x + threadIdx.x;
    if (idx >= MATELEMS) return;
    int mol = (int)(idx / (NORB * NORB));
    int rc = (int)(idx % (NORB * NORB));
    int r = rc / NORB, c = rc % NORB;
    float v = 0.0f;
    if (r == c) {
        int atom = mol * MS + r / 4;
        v = (float)(Z[atom] - 2) * 0.25f;
    }
    P[idx] = v;
}

// ---------------------------------------------------------------- Fock
__global__ void fock_pre_kernel(const float* __restrict__ P,
                                float* __restrict__ Ptot, float* __restrict__ vcoul)
{
    int a = blockIdx.x * blockDim.x + threadIdx.x;
    if (a >= NATOMS) return;
    int mol = a / MS, ai = a % MS;
    const float* Pm = P + (size_t)mol * NORB * NORB;
    float t = 0.0f;
#pragma unroll
    for (int u = 0; u < 4; ++u) t += Pm[(4 * ai + u) * NORB + 4 * ai + u];
    Ptot[a] = t;
    vcoul[a] = 0.0f;
}

__global__ void fock_pair_kernel(
    const int* __restrict__ idxi, const int* __restrict__ idxj,
    const int* __restrict__ pair_molid,
    const float* __restrict__ w_ss, const float* __restrict__ P,
    const float* __restrict__ Ptot, const float* __restrict__ Hcore,
    float* __restrict__ vcoul, float* __restrict__ F)
{
    int p = blockIdx.x * blockDim.x + threadIdx.x;
    if (p >= NPAIRS) return;
    int i = idxi[p], j = idxj[p];
    float wss = w_ss[p];
    atomicAdd(&vcoul[i], wss * Ptot[j]);
    atomicAdd(&vcoul[j], wss * Ptot[i]);
    int mol = pair_molid[p];
    int ai = i % MS, aj = j % MS;
    size_t mo = (size_t)mol * NORB * NORB;
    const float* Pm = P + mo;
    const float* Hm = Hcore + mo;
    float* Fm = F + mo;
#pragma unroll
    for (int u = 0; u < 4; ++u)
#pragma unroll
        for (int v = 0; v < 4; ++v) {
            float pp = Pm[(4 * ai + u) * NORB + 4 * aj + v];
            float fp = -0.5f * wss * pp;
            Fm[(4 * ai + u) * NORB + 4 * aj + v] = Hm[(4 * ai + u) * NORB + 4 * aj + v] + fp;
            Fm[(4 * aj + v) * NORB + 4 * ai + u] = Hm[(4 * aj + v) * NORB + 4 * ai + u] + fp;
        }
}

__global__ void fock_atom_kernel(
    const float* __restrict__ P, const float* __restrict__ vcoul,
    const float* __restrict__ Hcore,
    const float* __restrict__ g_ss, const float* __restrict__ g_sp,
    const float* __restrict__ g_pp, const float* __restrict__ g_p2,
    const float* __restrict__ h_sp, float* __restrict__ F)
{
    int a = blockIdx.x * blockDim.x + threadIdx.x;
    if (a >= NATOMS) return;
    int mol = a / MS, ai = a % MS;
    size_t mo = (size_t)mol * NORB * NORB;
    const float* Pm = P + mo;
    float Pd[4][4];
#pragma unroll
    for (int u = 0; u < 4; ++u)
#pragma unroll
        for (int v = 0; v < 4; ++v)
            Pd[u][v] = Pm[(4 * ai + u) * NORB + 4 * ai + v];
    float Pss = Pd[0][0];
    float Ppk[3] = { Pd[1][1], Pd[2][2], Pd[3][3] };
    float Ppp = Ppk[0] + Ppk[1] + Ppk[2];
    float gss = g_ss[a], gsp = g_sp[a], gpp = g_pp[a], gp2 = g_p2[a], hsp = h_sp[a];
    float csp = 1.5f * hsp - 0.5f * gsp;
    float fss = 0.5f * Pss * gss + Ppp * (gsp - 0.5f * hsp);
    float offc = 0.75f * gpp - 1.25f * gp2;
    float vc = vcoul[a];
    float Fd[4][4];
    Fd[0][0] = fss;
#pragma unroll
    for (int l = 0; l < 3; ++l) {
        Fd[0][1 + l] = Pd[0][1 + l] * csp;
        Fd[1 + l][0] = Pd[1 + l][0] * csp;
        float fpp = Pss * (gsp - 0.5f * hsp) + 0.5f * Ppk[l] * gpp +
                    (Ppp - Ppk[l]) * (1.25f * gp2 - 0.25f * gpp);
#pragma unroll
        for (int k = 0; k < 3; ++k)
            Fd[1 + k][1 + l] = (k == l) ? fpp : offc * Pd[1 + k][1 + l];
    }
#pragma unroll
    for (int u = 0; u < 4; ++u) Fd[u][u] += vc;
    const float* Hm = Hcore + mo;
    float* Fm = F + mo;
#pragma unroll
    for (int u = 0; u < 4; ++u)
#pragma unroll
        for (int v = 0; v < 4; ++v)
            Fm[(4 * ai + u) * NORB + 4 * ai + v] =
                Hm[(4 * ai + u) * NORB + 4 * ai + v] + Fd[u][v];
}

// ---------------------------------------------------------------- batched Jacobi eigh
// One workgroup (8 wave32s) per molecule. F staged by the Tensor Data Mover
// into a padded LDS image (row stride 195 dwords, odd -> conflict-free column
// walks); A and V both resident in the 320KB WGP LDS.
__global__ __launch_bounds__(256) void jacobi_eigh_kernel(
    const float* __restrict__ Fmat, const int* __restrict__ nocc,
    float* __restrict__ Vout, float* __restrict__ e_dest,
    float* __restrict__ occf)
{
    extern __shared__ float smem[];
    float* A = smem;                      // AELEMS (padded 192x192)
    float* V = A + AELEMS;                // AELEMS
    float* cs_c = V + AELEMS;             // 96
    float* cs_s = cs_c + 96;              // 96
    int* prm = (int*)(cs_s + 96);         // 192
    float* ev = (float*)(prm + 192);      // 192

    int mol = blockIdx.x;
    int tid = threadIdx.x;
    const float* Fm = Fmat + (size_t)mol * NORB * NORB;
    if (tid < 32)  // one wave issues the DMA descriptor
        tdm_load_f32_2d(Fm, (unsigned)(uintptr_t)A, NORB, NORB, NORB, 1);
    for (int idx = tid; idx < NORB * NORB; idx += blockDim.x) {
        int r = idx / NORB, c = idx % NORB;
        V[aoff(r, c)] = (r == c) ? 1.0f : 0.0f;
    }
    __builtin_amdgcn_s_wait_tensorcnt(0);
    __syncthreads();

    const int NR = NORB - 1;  // 191 rounds per sweep (round-robin ordering)
    for (int sweep = 0; sweep < JAC_SWEEPS; ++sweep) {
        for (int rnd = 0; rnd < NR; ++rnd) {
            if (tid < NORB / 2) {
                int i = tid;
                int p = (i == 0) ? 0 : (1 + (i - 1 + rnd) % NR);
                int jj = NORB - 1 - i;
                int q = 1 + (jj - 1 + rnd) % NR;
                if (p > q) { int t = p; p = q; q = t; }
                float app = A[aoff(p, p)], aqq = A[aoff(q, q)], apq = A[aoff(p, q)];
                float c = 1.0f, s = 0.0f;
                if (fabsf(apq) > 1e-12f) {
                    float theta = 0.5f * (aqq - app) / apq;
                    float t = copysignf(1.0f, theta) /
                              (fabsf(theta) + sqrtf(1.0f + theta * theta));
                    c = 1.0f / sqrtf(1.0f + t * t);
                    s = t * c;
                }
                cs_c[i] = c; cs_s[i] = s;
                prm[2 * i] = p; prm[2 * i + 1] = q;
            }
            __syncthreads();
            // column rotations on A and V (disjoint column pairs)
            for (int w = tid; w < (NORB / 2) * NORB; w += blockDim.x) {
                int pr = w / NORB, r = w % NORB;
                int p = prm[2 * pr], q = prm[2 * pr + 1];
                float c = cs_c[pr], s = cs_s[pr];
                int ip = aoff(r, p), iq = aoff(r, q);
                float ap = A[ip], aq = A[iq];
                A[ip] = c * ap - s * aq;
                A[iq] = s * ap + c * aq;
                float vp = V[ip], vq = V[iq];
                V[ip] = c * vp - s * vq;
                V[iq] = s * vp + c * vq;
            }
            __syncthreads();
            // row rotations on A (disjoint row pairs)
            for (int w = tid; w < (NORB / 2) * NORB; w += blockDim.x) {
                int pr = w / NORB, r = w % NORB;
                int p = prm[2 * pr], q = prm[2 * pr + 1];
                float c = cs_c[pr], s = cs_s[pr];
                int ip = aoff(p, r), iq = aoff(q, r);
                float ap = A[ip], aq = A[iq];
                A[ip] = c * ap - s * aq;
                A[iq] = s * ap + c * aq;
            }
            __syncthreads();
        }
    }
    if (tid < NORB) ev[tid] = A[aoff(tid, tid)];
    __syncthreads();
    if (tid < NORB) {
        float ek = ev[tid];
        int rank = 0;
        for (int k = 0; k < NORB; ++k) {
            float e2 = ev[k];
            rank += (e2 < ek) || (e2 == ek && k < tid);
        }
        e_dest[mol * NORB + rank] = ek;                    // ascending order
        occf[mol * NORB + tid] = (rank < nocc[mol]) ? 1.0f : 0.0f;
    }
    float* Vg = Vout + (size_t)mol * NORB * NORB;
    for (int idx = tid; idx < NORB * NORB; idx += blockDim.x)
        Vg[idx] = V[aoff(idx / NORB, idx % NORB)];
}

// ---------------------------------------------------------------- density: P = 2*Vocc*Vocc^T via WMMA f32
__global__ __launch_bounds__(256) void density_wmma_kernel(
    const float* __restrict__ V, const float* __restrict__ occf,
    float* __restrict__ Pnew)
{
    extern __shared__ float sm[];
    float* Vs = sm;  // AELEMS padded image of V
    int mol = blockIdx.x;
    int tid = threadIdx.x;
    const float* Vg = V + (size_t)mol * NORB * NORB;
    const float* of = occf + mol * NORB;
    if (tid < 32)
        tdm_load_f32_2d(Vg, (unsigned)(uintptr_t)Vs, NORB, NORB, NORB, 1);
    __builtin_amdgcn_s_wait_tensorcnt(0);
    __syncthreads();
    // occupancy-mask V in place (occ in {0,1}: mask once, use on both operands)
    for (int idx = tid; idx < NORB * NORB; idx += 256) {
        int r = idx / NORB, c = idx % NORB;
        Vs[aoff(r, c)] *= of[c];
    }
    __syncthreads();
    int wave = tid >> 5, lane = tid & 31;
    int lm = lane & 15;
    int khalf = (lane >> 4) << 1;  // lanes 0-15: K=0,1 ; lanes 16-31: K=2,3
    float* Pm = Pnew + (size_t)mol * NORB * NORB;
    for (int t = wave; t < 144; t += 8) {
        int i0 = (t / 12) * 16, j0 = (t % 12) * 16;
        v8f acc = {};
        // Pad contribution is constant inside each 64-wide K segment
        // (seg*65 = seg*64 elements + seg pad dwords), so the inner loop is
        // affine: immediate-offset ds_load_2addr pairs feeding v_wmma.
#pragma unroll
        for (int seg = 0; seg < 3; ++seg) {
            const float* Ai = Vs + ((i0 + lm) * AROW + seg * 65 + khalf);
            const float* Bj = Vs + ((j0 + lm) * AROW + seg * 65 + khalf);
#pragma unroll
            for (int k0 = 0; k0 < 64; k0 += 4) {
                v2f a, b;
                a.x = Ai[k0];
                a.y = Ai[k0 + 1];
                b.x = Bj[k0];                // B[k][n] = V[j0+n][k]
                b.y = Bj[k0 + 1];
                acc = __builtin_amdgcn_wmma_f32_16x16x4_f32(
                    false, a, false, b, (short)0, acc, false, false);
            }
        }
        int mb = (lane < 16) ? 0 : 8;
#pragma unroll
        for (int r = 0; r < 8; ++r)
            Pm[(i0 + mb + r) * NORB + j0 + lm] = 2.0f * acc[r];
    }
}

// ---------------------------------------------------------------- mix + err
__global__ void mix_kernel(float* __restrict__ P, const float* __restrict__ Pnew,
                           unsigned* __restrict__ errb, int do_err)
{
    size_t idx = (size_t)blockIdx.x * blockDim.x + threadIdx.x;
    if (idx >= MATELEMS) return;
    float po = P[idx], pn = Pnew[idx];
    if (do_err) {
        float d = fabsf(pn - po);
        atomicMax(&errb[idx / ((size_t)NORB * NORB)], __float_as_uint(d));
    }
    P[idx] = 0.5f * po + 0.5f * pn;
}

// ---------------------------------------------------------------- outputs
__global__ void charge_kernel(const int* __restrict__ Z, const float* __restrict__ P,
                              float* __restrict__ charge)
{
    int a = blockIdx.x * blockDim.x + threadIdx.x;
    if (a >= NATOMS) return;
    int mol = a / MS, ai = a % MS;
    const float* Pm = P + (size_t)mol * NORB * NORB;
    float t = 0.0f;
#pragma unroll
    for (int u = 0; u < 4; ++u) t += Pm[(4 * ai + u) * NORB + 4 * ai + u];
    charge[a] = (float)(Z[a] - 2) - t;
}

__global__ void notconv_kernel(const unsigned* __restrict__ errb, float* __restrict__ nc)
{
    int m = threadIdx.x;
    if (m < NMOLC) nc[m] = (__uint_as_float(errb[m]) > 1e-6f) ? 1.0f : 0.0f;
}

// ---------------------------------------------------------------- launch
extern "C" void kernel_launch(void* const* d_in, const int* in_sizes, int n_in,
                              void* d_out, int out_size, void* d_ws, size_t ws_size,
                              hipStream_t stream) {
    const int*   nocc   = (const int*)  d_in[3];
    const int*   Z      = (const int*)  d_in[4];
    const int*   atomid = (const int*)  d_in[7];
    const int*   pairid = (const int*)  d_in[8];
    const int*   idxi   = (const int*)  d_in[9];
    const int*   idxj   = (const int*)  d_in[10];
    const float* xij    = (const float*)d_in[13];
    const float* rij    = (const float*)d_in[14];
    const float* zeta_s = (const float*)d_in[15];
    const float* zeta_p = (const float*)d_in[16];
    const float* U_ss   = (const float*)d_in[17];
    const float* U_pp   = (const float*)d_in[18];
    const float* g_ss   = (const float*)d_in[19];
    const float* g_sp   = (const float*)d_in[20];
    const float* g_pp   = (const float*)d_in[21];
    const float* g_p2   = (const float*)d_in[22];
    const float* h_sp   = (const float*)d_in[23];
    const float* beta_s = (const float*)d_in[24];
    const float* beta_p = (const float*)d_in[25];
    (void)in_sizes; (void)n_in; (void)out_size; (void)ws_size;

    float* out   = (float*)d_out;
    float* Fo    = out;                 // 9437184
    float* e_o   = out + 9437184;       // 49152
    float* Po    = out + 9486336;       // 9437184 (live P)
    float* Hc    = out + 18923520;      // 9437184
    float* w_o   = out + 28360704;      // 28876800
    float* chg   = out + 57237504;      // 12288
    float* ncv   = out + 57249792;      // 256

    float* ws    = (float*)d_ws;
    float* w_ss  = ws;                  // 288768
    float* vatt  = ws + 288768;         // 12288
    float* vcoul = ws + 301056;         // 12288
    float* Ptot  = ws + 313344;         // 12288
    unsigned* errb = (unsigned*)(ws + 325632); // 256
    float* occf  = ws + 325888;         // 49152
    float* e_scr = ws + 375040;         // 49152
    float* Pnew  = ws + 424192;         // 9437184
    float* Vm    = ws + 9861376;        // 9437184

    const int ATOM_BLK = NATOMS / 256;          // 48
    const int PAIR_BLK = NPAIRS / 256;          // 1128
    const int ELEM_BLK = (int)(MATELEMS / 256); // 36864
    const size_t SMEM_J = (size_t)(2 * AELEMS + 96 + 96 + 192 + 192) * 4; // ~295 KB
    const size_t SMEM_D = (size_t)AELEMS * 4;                             // ~147 KB

    zero_f32_kernel<<<ATOM_BLK, 256, 0, stream>>>(vatt, NATOMS);
    zero_f32_kernel<<<1, 256, 0, stream>>>((float*)errb, NMOLC);

    pair_hcore_kernel<<<PAIR_BLK, 256, 0, stream>>>(
        idxi, idxj, pairid, Z, xij, rij, zeta_s, zeta_p, g_ss,
        beta_s, beta_p, w_o, w_ss, vatt, Hc);
    hcore_diag_kernel<<<ATOM_BLK, 256, 0, stream>>>(Z, atomid, U_ss, U_pp, vatt, Hc);
    init_P_kernel<<<ELEM_BLK, 256, 0, stream>>>(Z, Po);

    for (int it = 0; it < SCF_ITERS; ++it) {
        fock_pre_kernel<<<ATOM_BLK, 256, 0, stream>>>(Po, Ptot, vcoul);
        fock_pair_kernel<<<PAIR_BLK, 256, 0, stream>>>(
            idxi, idxj, pairid, w_ss, Po, Ptot, Hc, vcoul, Fo);
        fock_atom_kernel<<<ATOM_BLK, 256, 0, stream>>>(
            Po, vcoul, Hc, g_ss, g_sp, g_pp, g_p2, h_sp, Fo);
        jacobi_eigh_kernel<<<NMOLC, 256, SMEM_J, stream>>>(Fo, nocc, Vm, e_scr, occf);
        density_wmma_kernel<<<NMOLC, 256, SMEM_D, stream>>>(Vm, occf, Pnew);
        mix_kernel<<<ELEM_BLK, 256, 0, stream>>>(Po, Pnew, errb, it == SCF_ITERS - 1);
    }

    // final F from mixed P, final eigenvalues, charge, convergence flags
    fock_pre_kernel<<<ATOM_BLK, 256, 0, stream>>>(Po, Ptot, vcoul);
    fock_pair_kernel<<<PAIR_BLK, 256, 0, stream>>>(
        idxi, idxj, pairid, w_ss, Po, Ptot, Hc, vcoul, Fo);
    fock_atom_kernel<<<ATOM_BLK, 256, 0, stream>>>(
        Po, vcoul, Hc, g_ss, g_sp, g_pp, g_p2, h_sp, Fo);
    jacobi_eigh_kernel<<<NMOLC, 256, SMEM_J, stream>>>(Fo, nocc, Vm, e_o, occf);
    charge_kernel<<<ATOM_BLK, 256, 0, stream>>>(Z, Po, chg);
    notconv_kernel<<<1, 256, 0, stream>>>(errb, ncv);
}